// DigitCaps_14104672600339
// MI455X (gfx1250) — compile-verified
//
#include <hip/hip_runtime.h>
#include <math.h>

// DigitCaps routing for MI455X (gfx1250, wave32, WMMA), v2.
//
// Shapes: B=64, I=2048, Q=16, J=32, P=32.  hat[b,j,i,p] = sum_q in[b,i,q]*W[j,i,p,q]
// Never materialize hat (537MB). Three fused passes; pass(vin):
//   per (b,i): d[j] = vin[b,j,:].hat[b,j,i,:]; c = softmax_j(d); s[b,j,:] += c[j]*hat
// pass1 vin=0 (softmax(0)=1/J exactly), pass2 vin=v1, pass3 vin=v1+v2
// (routing logits are linear in v.hat with identical hat).
// Each pass reads W (134MB, fits the 192MB L2 so HBM sees it ~once) + inputs (8MB).
// hat is computed per-i as a 64x16x1024 fp32 GEMM via V_WMMA_F32_16X16X4_F32.
//
// v2: workgroup handles HALF the batch (32 b) to kill the v1 register spills:
//   acc 128->64 regs/thread, hat LDS 256KB->128KB (total 138KB => 2 WG/WGP possible).
//   B fragments loaded as direct v2f (no over-fetch + cndmask selects).

#define B_   64
#define I_   2048
#define Q_   16
#define J_   32
#define P_   32
#define JP_  1024
#define BH_  32           // batch rows per workgroup (half of B)
#define EPSQ 1e-7f
#define G_   512          // (256 i-chunks) x (2 b-halves)
#define IPW  8            // i-values per workgroup
#define NT   512          // 16 waves of 32

typedef float v2f __attribute__((ext_vector_type(2)));
typedef float v4f __attribute__((ext_vector_type(4)));
typedef float v8f __attribute__((ext_vector_type(8)));

// LDS plan (floats): hat[32*1024] | inp[32*16] | dbuf[32*32] | cbuf[32*32]
#define LDS_HAT   0
#define LDS_INP   32768
#define LDS_D     33280
#define LDS_C     34304
#define LDS_TOTAL 35328   // * 4B = 141,312 bytes (2 workgroups fit in 320KB WGP LDS)

__global__ __launch_bounds__(NT) __attribute__((amdgpu_waves_per_eu(4)))
void caps_pass_kernel(
    const float* __restrict__ inputs,   // [B][I][Q]
    const float* __restrict__ W,        // [J][I][P][Q]
    const float* __restrict__ vin,      // [B][J][P]
    float*       __restrict__ s)        // [B][J][P], pre-zeroed; atomic accumulated
{
  extern __shared__ float lds[];
  float* hat  = lds + LDS_HAT;
  float* inp  = lds + LDS_INP;
  float* dbuf = lds + LDS_D;
  float* cbuf = lds + LDS_C;

  const int t    = threadIdx.x;
  const int wave = t >> 5;       // 0..15
  const int lane = t & 31;
  const int hi   = lane >> 4;    // half-wave selects K-pair (ISA 7.12.2 f32 A/B layout)
  const int l15  = lane & 15;

  const int half  = blockIdx.x & 1;      // which 32-batch half
  const int bbase = half * BH_;
  const int ichunk = blockIdx.x >> 1;    // 0..255

  // Persistent output accumulators: thread t owns local b = t>>4, jp lanes with
  // jp%16 == t&15; jp rotated by b to stagger LDS banks.
  const int ab    = t >> 4;      // local b, 0..31
  const int aslot = t & 15;
  float acc[64];
#pragma unroll
  for (int k = 0; k < 64; ++k) acc[k] = 0.0f;

  for (int ii = 0; ii < IPW; ++ii) {
    const int i = ichunk * IPW + ii;

    // ---- stage inputs[bbase:bbase+32, i, :] (32x16 f32 = 2KB) into LDS ----
    inp[t] = inputs[(size_t)(bbase + (t >> 4)) * (I_ * Q_) + (size_t)i * Q_ + (t & 15)];
    __syncthreads();

    // ---- A fragments (16x4 f32) from LDS: v0=A[m][4k+2*hi], v1=A[m][4k+2*hi+1] ----
    v2f afrag[2][4];
#pragma unroll
    for (int m = 0; m < 2; ++m)
#pragma unroll
      for (int k = 0; k < 4; ++k)
        afrag[m][k] = *(const v2f*)(inp + (m * 16 + l15) * Q_ + 4 * k + 2 * hi);

    // ---- GEMM: wave owns ntiles [4w,4w+4); 2 mtiles; K=16 via 4 chained WMMAs ----
#pragma unroll
    for (int n = 0; n < 4; ++n) {
      const int ntile = wave * 4 + n;
      const int jp0 = ntile * 16 + l15;
      const int j = jp0 >> 5, p = jp0 & 31;
      const size_t base = ((size_t)(j * I_ + i) * P_ + p) * Q_ + 2 * hi;
      v2f bf[4];
#pragma unroll
      for (int k = 0; k < 4; ++k)       // B: v0=B[K=2hi][n], v1=B[K=2hi+1][n]
        bf[k] = *(const v2f*)(W + base + 4 * k);
#pragma unroll
      for (int m = 0; m < 2; ++m) {
        v8f d = {};
#pragma unroll
        for (int k = 0; k < 4; ++k)
          d = __builtin_amdgcn_wmma_f32_16x16x4_f32(
                  false, afrag[m][k], false, bf[k], (short)0, d, false, false);
#pragma unroll
        for (int r = 0; r < 8; ++r)     // D layout: lane=N col, VGPR r = M row
          hat[(m * 16 + hi * 8 + r) * JP_ + jp0] = d[r];
      }
    }
    __syncthreads();

    // ---- d[lb][j] = vin[bbase+lb, j, :] . hat[lb, j*32 : j*32+32] ----
#pragma unroll
    for (int u = 0; u < 2; ++u) {
      const int idx = t + NT * u;                // 1024 = 32*32 (lb,j) pairs
      const int lb = idx >> 5, j = idx & 31;
      const float* hr = hat + lb * JP_ + j * 32;
      const v4f*   vr = (const v4f*)(vin + ((size_t)(bbase + lb) * J_ + j) * P_);
      float sum = 0.0f;
#pragma unroll
      for (int p4 = 0; p4 < 8; ++p4) {
        v4f h = *(const v4f*)(hr + 4 * p4);
        v4f v = vr[p4];
        sum += h.x * v.x + h.y * v.y + h.z * v.z + h.w * v.w;
      }
      dbuf[idx] = sum;
    }
    __syncthreads();

    // ---- softmax over j (32 values) per local b; threads 0..31 ----
    if (t < BH_) {
      const float* dr = dbuf + t * J_;
      float mx = dr[0];
#pragma unroll
      for (int j = 1; j < J_; ++j) mx = fmaxf(mx, dr[j]);
      float sum = 0.0f;
#pragma unroll
      for (int j = 0; j < J_; ++j) sum += __expf(dr[j] - mx);
      const float inv = 1.0f / sum;
#pragma unroll
      for (int j = 0; j < J_; ++j) cbuf[t * J_ + j] = __expf(dr[j] - mx) * inv;
    }
    __syncthreads();

    // ---- accumulate s_partial[lb][jp] += c[lb][j]*hat[lb][jp] in registers ----
#pragma unroll
    for (int k = 0; k < 64; ++k) {
      const int jp = ((ab + k) & 63) * 16 + aslot;
      acc[k] += cbuf[ab * J_ + (jp >> 5)] * hat[ab * JP_ + jp];
    }
    __syncthreads();   // protect hat/inp before next i iteration
  }

  // ---- flush partials: hardware fp32 global atomics (s pre-zeroed) ----
#pragma unroll
  for (int k = 0; k < 64; ++k) {
    const int jp = ((ab + k) & 63) * 16 + aslot;
    unsafeAtomicAdd(s + (size_t)(bbase + ab) * JP_ + jp, acc[k]);
  }
}

__global__ void zero_kernel(float* __restrict__ p, int n) {
  int i = blockIdx.x * blockDim.x + threadIdx.x;
  if (i < n) p[i] = 0.0f;
}

// mode 0: out0 = squash(s); out1 = squash(s)        (v1 and vsum init)
// mode 1: out0 += squash(s)                          (vsum += v2)
// mode 2: out0 = squash(s)                           (final output)
__global__ void squash_kernel(const float* __restrict__ s,
                              float* __restrict__ out0,
                              float* __restrict__ out1, int mode) {
  int idx = blockIdx.x * blockDim.x + threadIdx.x;   // one (b,j) per thread
  if (idx >= B_ * J_) return;
  const float* v = s + (size_t)idx * P_;
  float s2 = 0.0f;
#pragma unroll
  for (int p = 0; p < P_; ++p) s2 += v[p] * v[p];
  const float scale = s2 / (1.0f + s2) / sqrtf(s2 + EPSQ);
#pragma unroll
  for (int p = 0; p < P_; ++p) {
    const float q = scale * v[p];
    const size_t o = (size_t)idx * P_ + p;
    if (mode == 0)      { out0[o] = q; out1[o] = q; }
    else if (mode == 1) { out0[o] += q; }
    else                { out0[o] = q; }
  }
}

extern "C" void kernel_launch(void* const* d_in, const int* in_sizes, int n_in,
                              void* d_out, int out_size, void* d_ws, size_t ws_size,
                              hipStream_t stream) {
  const float* inputs = (const float*)d_in[0];   // [64][2048][16]
  const float* W      = (const float*)d_in[1];   // [32][2048][32][16]
  float* out  = (float*)d_out;                   // [64][32][32]

  // workspace: s | v1 | vsum   (3 * 65536 floats = 768KB)
  float* s    = (float*)d_ws;
  float* v1   = s  + (size_t)B_ * J_ * P_;
  float* vsum = v1 + (size_t)B_ * J_ * P_;
  const int NV = B_ * J_ * P_;                   // 65536
  const size_t ldsBytes = (size_t)LDS_TOTAL * sizeof(float);

  // pass 1: vin = 0  (softmax(0) = 1/J, exactly the reference's first round)
  zero_kernel<<<(2 * NV + 255) / 256, 256, 0, stream>>>(s, 2 * NV);   // zeros s and v1
  caps_pass_kernel<<<G_, NT, ldsBytes, stream>>>(inputs, W, v1, s);
  squash_kernel<<<(B_ * J_ + 255) / 256, 256, 0, stream>>>(s, v1, vsum, 0);

  // pass 2: vin = v1
  zero_kernel<<<(NV + 255) / 256, 256, 0, stream>>>(s, NV);
  caps_pass_kernel<<<G_, NT, ldsBytes, stream>>>(inputs, W, v1, s);
  squash_kernel<<<(B_ * J_ + 255) / 256, 256, 0, stream>>>(s, vsum, nullptr, 1);

  // pass 3: vin = v1 + v2  (logits are linear in v.hat) -> final squash to d_out
  zero_kernel<<<(NV + 255) / 256, 256, 0, stream>>>(s, NV);
  caps_pass_kernel<<<G_, NT, ldsBytes, stream>>>(inputs, W, vsum, s);
  squash_kernel<<<(B_ * J_ + 255) / 256, 256, 0, stream>>>(s, out, nullptr, 2);
}